// Gate_403726925997
// MI455X (gfx1250) — compile-verified
//
#include <hip/hip_runtime.h>
#include <hip/hip_bf16.h>
#include <math.h>

typedef __attribute__((ext_vector_type(2))) float v2f;
typedef __attribute__((ext_vector_type(8))) float v8f;

#define D_DIM 2048
#define E_DIM 64
#define KC    64
#define NCHUNK (D_DIM / KC)        // 32
#define ROWS_PER_BLOCK 128
#define XPAD  68   // 64 + 4 pad -> conflict-free ds_load_b64, keeps 16B alignment
#define WPAD  68
#define LPAD  65   // logits row stride: lane+e rotates across banks

// --- CDNA5 async global->LDS copy (ASYNCcnt-tracked, bypasses VGPRs) ---
__device__ __forceinline__ void async_b128_to_lds(float* lds_ptr, const float* gptr) {
    // flat->LDS mapping: low 32 bits of the generic LDS pointer are the DS offset
    unsigned int lds_off = (unsigned int)(unsigned long long)lds_ptr;
    unsigned long long ga = (unsigned long long)gptr;
    asm volatile("global_load_async_to_lds_b128 %0, %1, off"
                 :: "v"(lds_off), "v"(ga) : "memory");
}

#if defined(__has_builtin) && __has_builtin(__builtin_amdgcn_s_wait_asynccnt)
#define WAIT_ASYNC(n) __builtin_amdgcn_s_wait_asynccnt(n)
#else
#define WAIT_ASYNC(n) asm volatile("s_wait_asynccnt %0" :: "i"(n) : "memory")
#endif

__global__ __launch_bounds__(256) void moe_gate_kernel(
    const float* __restrict__ x, const float* __restrict__ w_gate,
    const float* __restrict__ bias, float* __restrict__ out, int ntok)
{
    __shared__ float ldsx[2][ROWS_PER_BLOCK * XPAD];   // ~69.6 KB
    __shared__ float ldsw[2][E_DIM * WPAD];            // ~34.8 KB
    __shared__ float ldsb[E_DIM];
    // logits region aliases x-buffer 0 (free after the last buffer-0 compute,
    // which is separated from the epilogue by a barrier; spill/readback are wave-local)
    float* ldsl = &ldsx[0][0];                          // needs 128*65 <= 128*68 floats

    const int tid    = threadIdx.x;
    const int wv     = tid >> 5;        // wave id 0..7
    const int lane   = tid & 31;
    const int lane16 = lane & 15;
    const int hi     = lane >> 4;       // half-wave select
    const int row0   = blockIdx.x * ROWS_PER_BLOCK;

    if (tid < E_DIM) ldsb[tid] = bias[tid];

    v8f acc0 = {0,0,0,0,0,0,0,0};
    v8f acc1 = {0,0,0,0,0,0,0,0};
    v8f acc2 = {0,0,0,0,0,0,0,0};
    v8f acc3 = {0,0,0,0,0,0,0,0};

    const int lr = tid >> 4;          // 0..15 loader row
    const int lc = (tid & 15) << 2;   // 0..60 loader col (b128 granules)

    // ---- stage one K-chunk (12 async b128 per thread: 8 for x, 4 for w) ----
    auto stage = [&](int k0, int buf) {
        float* bx = &ldsx[buf][0];
        float* bw = &ldsw[buf][0];
        #pragma unroll
        for (int p = 0; p < 8; ++p) {
            int r = p * 16 + lr;
            async_b128_to_lds(bx + r * XPAD + lc,
                              x + (size_t)(row0 + r) * D_DIM + k0 + lc);
        }
        #pragma unroll
        for (int p = 0; p < 4; ++p) {
            int e = p * 16 + lr;
            async_b128_to_lds(bw + e * WPAD + lc,
                              w_gate + (size_t)e * D_DIM + k0 + lc);
        }
    };

    // ---- prologue: fill both buffers ----
    stage(0, 0);            // ASYNCcnt = 12
    stage(KC, 1);           // ASYNCcnt = 24
    WAIT_ASYNC(12);         // chunk 0 landed (async loads complete in order)
    __syncthreads();

    for (int c = 0; c < NCHUNK; ++c) {
        // ---- compute chunk c from buffer c&1 ----
        const float* bx = &ldsx[c & 1][0];
        const float* bw = &ldsw[c & 1][0];
        const float* ax  = bx + (wv * 16 + lane16) * XPAD + (hi << 1);
        const float* bwp = bw + lane16 * WPAD + (hi << 1);
        #pragma unroll 4
        for (int kk = 0; kk < KC; kk += 4) {
            v2f a  = *(const v2f*)(ax + kk);
            v2f b0 = *(const v2f*)(bwp + 0 * 16 * WPAD + kk);
            v2f b1 = *(const v2f*)(bwp + 1 * 16 * WPAD + kk);
            v2f b2 = *(const v2f*)(bwp + 2 * 16 * WPAD + kk);
            v2f b3 = *(const v2f*)(bwp + 3 * 16 * WPAD + kk);
            acc0 = __builtin_amdgcn_wmma_f32_16x16x4_f32(false, a, false, b0, (short)0, acc0, false, false);
            acc1 = __builtin_amdgcn_wmma_f32_16x16x4_f32(false, a, false, b1, (short)0, acc1, false, false);
            acc2 = __builtin_amdgcn_wmma_f32_16x16x4_f32(false, a, false, b2, (short)0, acc2, false, false);
            acc3 = __builtin_amdgcn_wmma_f32_16x16x4_f32(false, a, false, b3, (short)0, acc3, false, false);
        }
        __syncthreads();                 // all waves done reading buffer c&1

        if (c + 2 < NCHUNK) {
            stage((c + 2) * KC, c & 1);  // refill the buffer just drained
            WAIT_ASYNC(12);              // chunk c+1 landed
        } else {
            WAIT_ASYNC(0);               // drain tail
        }
        __syncthreads();                 // chunk c+1 visible block-wide
    }

    // ---- spill logits: C/D layout VGPR r -> M = r + 8*hi, N = 16*t + lane16 ----
    {
        float* lrow = ldsl + lane16;
        #pragma unroll
        for (int r = 0; r < 8; ++r) {
            int m = wv * 16 + r + 8 * hi;
            lrow[m * LPAD +  0] = acc0[r];
            lrow[m * LPAD + 16] = acc1[r];
            lrow[m * LPAD + 32] = acc2[r];
            lrow[m * LPAD + 48] = acc3[r];
        }
    }
    // wave-local spill/readback: intra-wave DS ordering via dscnt waits

    if (lane < 16) {
        const int rl    = wv * 16 + lane;
        const int token = row0 + rl;
        const float* lg = ldsl + rl * LPAD;

        float bv[8]; int bi[8];
        #pragma unroll
        for (int j = 0; j < 8; ++j) { bv[j] = -INFINITY; bi[j] = 0; }

        // branch-free sorted top-8 insert; strict '>' keeps earliest index on ties
        for (int e = 0; e < E_DIM; ++e) {
            float v  = lg[e] + ldsb[e];
            int   ix = e;
            #pragma unroll
            for (int j = 0; j < 8; ++j) {
                bool gt = v > bv[j];
                float tv = gt ? bv[j] : v;
                int   ti = gt ? bi[j] : ix;
                bv[j] = gt ? v  : bv[j];
                bi[j] = gt ? ix : bi[j];
                v = tv; ix = ti;
            }
        }

        float sw[8]; float s = 0.f;
        #pragma unroll
        for (int j = 0; j < 8; ++j) {
            float raw = lg[bi[j]];              // sigmoid of UNbiased logit
            float sg  = 1.0f / (1.0f + __expf(-raw));
            sw[j] = sg; s += sg;
        }
        float inv = 1.0f / s;

        float* outw = out + (size_t)token * 8;
        int*   outi = (int*)out + (size_t)ntok * 8 + (size_t)token * 8;
        #pragma unroll
        for (int j = 0; j < 8; ++j) {
            outw[j] = sw[j] * inv;
            outi[j] = bi[j];
        }
    }
}

extern "C" void kernel_launch(void* const* d_in, const int* in_sizes, int n_in,
                              void* d_out, int out_size, void* d_ws, size_t ws_size,
                              hipStream_t stream) {
    const float* x    = (const float*)d_in[0];
    const float* w    = (const float*)d_in[1];
    const float* bias = (const float*)d_in[2];
    float* out = (float*)d_out;

    const int E = in_sizes[2];               // 64
    const int D = in_sizes[1] / E;           // 2048
    const int ntok = in_sizes[0] / D;        // 16384
    const int nblocks = ntok / ROWS_PER_BLOCK;

    moe_gate_kernel<<<nblocks, 256, 0, stream>>>(x, w, bias, out, ntok);
}